// G_LSTM_layer_22634477650082
// MI455X (gfx1250) — compile-verified
//
#include <hip/hip_runtime.h>
#include <math.h>

// Problem constants: B=8, C=32, N=512, T=12, 8 heads.
#define BB 8
#define CC 32
#define NN 512
#define TT 12

typedef float v2f __attribute__((ext_vector_type(2)));
typedef float v8f __attribute__((ext_vector_type(8)));

__device__ __forceinline__ float elu1(float x) { return x > 0.0f ? x : expm1f(x); }
__device__ __forceinline__ float sigm(float x) { return 1.0f / (1.0f + expf(-x)); }

// ---------------------------------------------------------------------------
// Kernel 1: precompute xs[t][b*32+i] = sum_n input[b,i,n,t], and the initial
// cell partial sums partial[2j]+partial[2j+1] = sum_n cell[b,i,n].
// One block per (b,i) row: 6144 contiguous floats of input staged via LDS.
// ---------------------------------------------------------------------------
__global__ __launch_bounds__(256) void init_kernel(const float* __restrict__ input,
                                                   const float* __restrict__ cell,
                                                   float* __restrict__ xs_ws,
                                                   float* __restrict__ partial) {
  const int j = blockIdx.x;      // 0..255  (= b*32 + i)
  const int tid = threadIdx.x;
  __shared__ float buf[NN * TT]; // 6144 floats = 24 KB
  __shared__ float red[256];
  __shared__ float part2[192];

  const float* src = input + (size_t)j * (NN * TT);
  for (int k = tid; k < NN * TT; k += 256) buf[k] = src[k];

  // initial cell row reduction (512 contiguous floats)
  red[tid] = cell[(size_t)j * NN + tid] + cell[(size_t)j * NN + 256 + tid];
  __syncthreads();
  for (int s = 128; s > 0; s >>= 1) {
    if (tid < s) red[tid] += red[tid + s];
    __syncthreads();
  }
  if (tid == 0) { partial[2 * j] = red[0]; partial[2 * j + 1] = 0.0f; }

  // per-timestep reductions over n (buf[n*12 + t])
  if (tid < 192) {
    const int t = tid / 16, l = tid % 16;
    float s = 0.0f;
    for (int k = 0; k < 32; ++k) s += buf[(l + 16 * k) * TT + t];
    part2[tid] = s;
  }
  __syncthreads();
  if (tid < TT) {
    float s = 0.0f;
    for (int l = 0; l < 16; ++l) s += part2[tid * 16 + l];
    xs_ws[tid * 256 + j] = s;
  }
}

// ---------------------------------------------------------------------------
// Kernel 2 (per timestep): 8 waves = 8 heads. Two fp32 WMMA GEMM layers:
//   s1 = in @ W1  (in: 8x32 zero-padded to 16x32), u = 512*elu(elu(s1))
//   s2 = u  @ W2,  out = elu(elu(elu(s2)))
// All LDS operands are padded to 16 rows (rows 8-15 zero) so every fragment
// load/store is unconditional -> no EXEC save/restore around the WMMA chain.
// ---------------------------------------------------------------------------
__global__ __launch_bounds__(256) void head_kernel(const float* __restrict__ w1,
                                                   const float* __restrict__ w2,
                                                   const float* __restrict__ xs_ws,
                                                   const float* __restrict__ partial,
                                                   float* __restrict__ outb,
                                                   int t) {
  __shared__ float inb[2][16 * CC];      // padded 16x32, rows 8-15 = 0
  __shared__ float hsh[8][16][CC];       // padded per-head hidden (16 KB)
  const int tid = threadIdx.x;

  // rows 0-7: data; rows 8-15: zeros
  inb[1][tid] = partial[2 * tid] + partial[2 * tid + 1];  // cs[b,i]
  inb[0][tid] = xs_ws[t * 256 + tid];                     // xs[b,i]
  inb[0][256 + tid] = 0.0f;
  inb[1][256 + tid] = 0.0f;
  __syncthreads();

  const int h = tid >> 5;
  const int lane = tid & 31;
  const int sel = (h >= 4) ? 1 : 0;
  const float* w1p = w1 + ((size_t)t * 8 + h) * (CC * CC);
  const float* w2p = w2 + ((size_t)t * 8 + h) * (CC * CC);

  const int m = lane & 15;                 // A row / B,D column-within-tile
  const int khi = (lane >> 4) << 1;        // lanes 16-31 carry K+2/K+3
  const int rhi = (lane >> 4) << 3;        // D rows: lanes 16-31 -> M = r+8

  const v8f zero = {0.f, 0.f, 0.f, 0.f, 0.f, 0.f, 0.f, 0.f};
  v8f acc0 = zero, acc1 = zero;

  // ---- layer 1: s1 = in @ W1 ----
#pragma unroll
  for (int k0 = 0; k0 < 8; ++k0) {
    const int kk = k0 * 4 + khi;
    const v2f a = *(const v2f*)&inb[sel][m * CC + kk];   // ds_load_b64
    v2f b0, b1;
    b0.x = w1p[kk * CC + m];            b0.y = w1p[(kk + 1) * CC + m];
    b1.x = w1p[kk * CC + m + 16];       b1.y = w1p[(kk + 1) * CC + m + 16];
    acc0 = __builtin_amdgcn_wmma_f32_16x16x4_f32(false, a, false, b0, (short)0, acc0, false, false);
    acc1 = __builtin_amdgcn_wmma_f32_16x16x4_f32(false, a, false, b1, (short)0, acc1, false, false);
  }

  // hs = N * elu(elu(s1)); lanes>=16 hold rows 8-15 which are exactly 0,
  // and elu(0)=0, so all lanes store -> padding rows written naturally.
#pragma unroll
  for (int r = 0; r < BB; ++r) {
    hsh[h][r + rhi][m]      = 512.0f * elu1(elu1(acc0[r]));
    hsh[h][r + rhi][m + 16] = 512.0f * elu1(elu1(acc1[r]));
  }
  __syncthreads();

  // ---- layer 2: s2 = hs @ W2 ----
  acc0 = zero; acc1 = zero;
#pragma unroll
  for (int k0 = 0; k0 < 8; ++k0) {
    const int kk = k0 * 4 + khi;
    const v2f a = *(const v2f*)&hsh[h][m][kk];           // ds_load_b64
    v2f b0, b1;
    b0.x = w2p[kk * CC + m];            b0.y = w2p[(kk + 1) * CC + m];
    b1.x = w2p[kk * CC + m + 16];       b1.y = w2p[(kk + 1) * CC + m + 16];
    acc0 = __builtin_amdgcn_wmma_f32_16x16x4_f32(false, a, false, b0, (short)0, acc0, false, false);
    acc1 = __builtin_amdgcn_wmma_f32_16x16x4_f32(false, a, false, b1, (short)0, acc1, false, false);
  }

  // outb padded to 16 rows per head; rows 8-15 receive elu^3(0)=0.
#pragma unroll
  for (int r = 0; r < BB; ++r) {
    outb[h * 512 + (r + rhi) * CC + m]      = elu1(elu1(elu1(acc0[r])));
    outb[h * 512 + (r + rhi) * CC + m + 16] = elu1(elu1(elu1(acc1[r])));
  }
}

// ---------------------------------------------------------------------------
// Kernel 3 (per timestep): LSTM gates over (b,c,n). Block-uniform (b,c) so the
// 8 head outputs become scalar loads. Writes H[...,t], updates c, and emits a
// deterministic per-block partial sum of c_new for the next step's cs.
// ---------------------------------------------------------------------------
__global__ __launch_bounds__(256) void gate_kernel(const float* __restrict__ outb,
                                                   const float* __restrict__ bias,
                                                   const float* __restrict__ cin,
                                                   float* __restrict__ cout,
                                                   float* __restrict__ H,
                                                   float* __restrict__ partial,
                                                   int t) {
  const int tid = threadIdx.x;
  const int bc = blockIdx.x >> 1;                 // b*32 + c
  const int n = ((blockIdx.x & 1) << 8) + tid;
  const int cch = bc & 31;

  const float g0 = outb[0 * 512 + bc] + outb[4 * 512 + bc];
  const float g1 = outb[1 * 512 + bc] + outb[5 * 512 + bc];
  const float g2 = outb[2 * 512 + bc] + outb[6 * 512 + bc];
  const float g3 = outb[3 * 512 + bc] + outb[7 * 512 + bc];

  const size_t bstride = (size_t)CC * NN * TT;    // per-gate stride in bias
  const size_t bidx = ((size_t)cch * NN + n) * TT + t;
  const float ig = sigm(g0 + bias[bidx]);
  const float fg = sigm(g1 + bias[bidx + bstride]);
  const float og = sigm(g2 + bias[bidx + 2 * bstride]);
  const float ct = tanhf(g3 + bias[bidx + 3 * bstride]);

  const float cold = cin[(size_t)bc * NN + n];
  const float cn = fg * cold + ig * ct;
  H[((size_t)bc * NN + n) * TT + t] = og * tanhf(cn);
  cout[(size_t)bc * NN + n] = cn;

  __shared__ float red[256];
  red[tid] = cn;
  __syncthreads();
  for (int s = 128; s > 0; s >>= 1) {
    if (tid < s) red[tid] += red[tid + s];
    __syncthreads();
  }
  if (tid == 0) partial[blockIdx.x] = red[0];
}

// ---------------------------------------------------------------------------
extern "C" void kernel_launch(void* const* d_in, const int* in_sizes, int n_in,
                              void* d_out, int out_size, void* d_ws, size_t ws_size,
                              hipStream_t stream) {
  const float* input = (const float*)d_in[0]; // (B,C,N,T)
  const float* cell  = (const float*)d_in[1]; // (B,C,N,1)
  // d_in[2] = adj  : mathematically dead (softmax rows sum to 1)
  const float* w1    = (const float*)d_in[3]; // (T,8,C,C)
  // d_in[4] = a1   : dead
  const float* w2    = (const float*)d_in[5]; // (T,8,C,C)
  // d_in[6] = a2   : dead
  const float* bias  = (const float*)d_in[7]; // (4,C,N,T)

  float* out  = (float*)d_out;                     // (B,C,N,T) flat
  float* cfin = out + (size_t)BB * CC * NN * TT;   // (B,C,N,1) flat

  float* ws      = (float*)d_ws;
  float* xs_ws   = ws;               // 12*256 = 3072
  float* partial = ws + 3072;        // 512
  float* outb    = ws + 3584;        // 8*16*32 = 4096 (rows 8-15 padding)
  float* cbuf    = ws + 7680;        // 8*32*512 = 131072

  init_kernel<<<256, 256, 0, stream>>>(input, cell, xs_ws, partial);

  for (int t = 0; t < TT; ++t) {
    head_kernel<<<1, 256, 0, stream>>>(w1, w2, xs_ws, partial, outb, t);
    const float* cin = (t == 0) ? cell : cbuf;
    float* cout = (t == TT - 1) ? cfin : cbuf;
    gate_kernel<<<512, 256, 0, stream>>>(outb, bias, cin, cout, out, partial, t);
  }
}